// SimCLR_36713380446721
// MI455X (gfx1250) — compile-verified
//
#include <hip/hip_runtime.h>
#include <math.h>

// SimCLR loss on MI455X (gfx1250, wave32).
//   total = sum_j LSE_col_j(S) - 2 * sum_r S[r, n+r],   S = zn zn^T / tau
// LSE shift = 1/tau = 1000 (provable column max: diagonal cos == 1).
// GEMM (8192x8192x128, 17.2 GFLOP) done with V_WMMA_F32_16X16X4_F32,
// streaming row tiles so S is never materialized. zn (4 MB) lives in L2.

typedef __attribute__((ext_vector_type(2))) float v2f;
typedef __attribute__((ext_vector_type(8))) float v8f;

#define NROWS 4096
#define DIM   128
#define NTOT  8192
#define TAU_INV 1000.0f

// ---------------------------------------------------------------- normalize
__global__ void __launch_bounds__(256)
simclr_normalize(const float* __restrict__ z1, const float* __restrict__ z2,
                 float* __restrict__ zn) {
  const int wave = threadIdx.x >> 5;
  const int lane = threadIdx.x & 31;
  const int row  = blockIdx.x * 8 + wave;
  const float* src = (row < NROWS) ? (z1 + (size_t)row * DIM)
                                   : (z2 + (size_t)(row - NROWS) * DIM);
  float4 v = ((const float4*)src)[lane];               // 128 = 32 lanes * 4
  float ss = v.x * v.x + v.y * v.y + v.z * v.z + v.w * v.w;
  #pragma unroll
  for (int off = 16; off; off >>= 1) ss += __shfl_xor(ss, off, 32);
  const float inv = 1.0f / fmaxf(sqrtf(ss), 1e-8f);
  float4 o; o.x = v.x * inv; o.y = v.y * inv; o.z = v.z * inv; o.w = v.w * inv;
  ((float4*)(zn + (size_t)row * DIM))[lane] = o;
}

// ------------------------------------------------- column sum-exp via WMMA
// One block = one 16-column strip of S. 4 waves split the 512 row tiles.
// A/B fragment layout for V_WMMA_F32_16X16X4_F32 (ISA 7.12.2): lane l holds
// matrix row/col (l&15), K-pair base (l>>4)*2 -> identical addressing for
// A (rows of zn) and B (columns of S == rows of zn).
__global__ void __launch_bounds__(128)
simclr_colsum(const float* __restrict__ zn, float* __restrict__ sumexp) {
  const int lane = threadIdx.x & 31;
  const int wave = threadIdx.x >> 5;
  const int mn   = lane & 15;       // row/col index within 16-tile
  const int kh   = lane >> 4;       // which K half-pair
  const int colBase = blockIdx.x * 16;

  // Preload B fragment for the whole K=128 (32 slabs of K=4) -> 64 VGPRs.
  v2f bfrag[32];
  const float* bptr = zn + (size_t)(colBase + mn) * DIM + kh * 2;
  #pragma unroll
  for (int kk = 0; kk < 32; ++kk)
    bfrag[kk] = *(const v2f*)(bptr + kk * 4);

  float colAcc = 0.0f;  // partial sum of exp(S - 1000) for column colBase+mn
  for (int rowTile = wave; rowTile < NTOT / 16; rowTile += 4) {
    const float* aptr = zn + (size_t)(rowTile * 16 + mn) * DIM + kh * 2;
    v8f c = {0.f, 0.f, 0.f, 0.f, 0.f, 0.f, 0.f, 0.f};
    #pragma unroll
    for (int kk = 0; kk < 32; ++kk) {
      v2f a = *(const v2f*)(aptr + kk * 4);
      c = __builtin_amdgcn_wmma_f32_16x16x4_f32(
              /*neg_a=*/false, a, /*neg_b=*/false, bfrag[kk],
              /*c_mod=*/(short)0, c, /*reuse_a=*/false, /*reuse_b=*/false);
    }
    // c[v] = dot(zn_row, zn_col); S = dot*1000; exp(S - 1000) = exp((dot-1)*1000)
    #pragma unroll
    for (int v = 0; v < 8; ++v)
      colAcc += __expf(fmaf(c[v], TAU_INV, -TAU_INV));
  }

  // fold the two lane-halves (C/D layout: lanes l and l^16 share column mn)
  colAcc += __shfl_xor(colAcc, 16, 32);

  __shared__ float cred[4][16];
  if (lane < 16) cred[wave][mn] = colAcc;
  __syncthreads();
  if (threadIdx.x < 16) {
    float s = cred[0][threadIdx.x] + cred[1][threadIdx.x] +
              cred[2][threadIdx.x] + cred[3][threadIdx.x];
    sumexp[colBase + threadIdx.x] = s;   // exclusive column ownership
  }
}

// ------------------------------------------------------- positive-pair dots
__global__ void __launch_bounds__(256)
simclr_pospair(const float* __restrict__ zn, float* __restrict__ posAcc) {
  const int wave = threadIdx.x >> 5;
  const int lane = threadIdx.x & 31;
  const int r = blockIdx.x * 8 + wave;
  const float4 a = ((const float4*)(zn + (size_t)r * DIM))[lane];
  const float4 b = ((const float4*)(zn + (size_t)(NROWS + r) * DIM))[lane];
  float d = a.x * b.x + a.y * b.y + a.z * b.z + a.w * b.w;
  #pragma unroll
  for (int off = 16; off; off >>= 1) d += __shfl_xor(d, off, 32);
  if (lane == 0) atomicAdd(posAcc, d * TAU_INV);   // S[r, n+r]
}

// ----------------------------------------------------------------- finalize
__global__ void __launch_bounds__(256)
simclr_finalize(const float* __restrict__ sumexp,
                const float* __restrict__ posAcc, float* __restrict__ out) {
  __shared__ double red[256];
  double acc = 0.0;
  for (int j = threadIdx.x; j < NTOT; j += 256)
    acc += 1000.0 + log((double)sumexp[j]);        // LSE_j with shift 1000
  red[threadIdx.x] = acc;
  __syncthreads();
  for (int s = 128; s; s >>= 1) {
    if (threadIdx.x < s) red[threadIdx.x] += red[threadIdx.x + s];
    __syncthreads();
  }
  if (threadIdx.x == 0) {
    double total = red[0] - 2.0 * (double)posAcc[0];
    out[0] = (float)(total / (double)(NROWS * 2)); // n*(k*k-k) = 8192
  }
}

// ------------------------------------------------------------------- launch
extern "C" void kernel_launch(void* const* d_in, const int* in_sizes, int n_in,
                              void* d_out, int out_size, void* d_ws, size_t ws_size,
                              hipStream_t stream) {
  (void)in_sizes; (void)n_in; (void)out_size; (void)ws_size;
  const float* z1 = (const float*)d_in[0];
  const float* z2 = (const float*)d_in[1];
  float* out = (float*)d_out;

  float* zn      = (float*)d_ws;            // 8192*128 f32 = 4 MB
  float* sumexp  = zn + (size_t)NTOT * DIM; // 8192 f32
  float* posAcc  = sumexp + NTOT;           // 1 f32

  hipMemsetAsync(posAcc, 0, sizeof(float), stream);  // graph-capturable
  simclr_normalize<<<NTOT / 8, 256, 0, stream>>>(z1, z2, zn);
  simclr_colsum  <<<NTOT / 16, 128, 0, stream>>>(zn, sumexp);
  simclr_pospair <<<NROWS / 8, 256, 0, stream>>>(zn, posAcc);
  simclr_finalize<<<1, 256, 0, stream>>>(sumexp, posAcc, out);
}